// GCN_4037269259073
// MI455X (gfx1250) — compile-verified
//
#include <hip/hip_runtime.h>
#include <hip/hip_bf16.h>

typedef float v2f __attribute__((ext_vector_type(2)));
typedef float v8f __attribute__((ext_vector_type(8)));

#define NN 100000          // nodes
#define NE 1600000         // edges
#define GDIM 128
#define H1 64
#define H2 64
#define NREL 8
#define NBASES 30
#define NCAT1 (H1 + NREL * H1)   // 576 = root(64) + 8 relations * 64
#define NCAT2 (H2 + H2)          // 128 = w_rel(64) + w_root(64)

// ---------------- weight preparation ----------------

// Wcat1[128][576]: cols 0..63 = root, cols 64+64r.. = sum_b comp[r][b]*basis[b][i][o]
__global__ __launch_bounds__(256) void build_wcat1(
    const float* __restrict__ basis, const float* __restrict__ comp,
    const float* __restrict__ root, float* __restrict__ W) {
  int idx = blockIdx.x * blockDim.x + threadIdx.x;
  if (idx >= GDIM * NCAT1) return;
  int i = idx / NCAT1;
  int o = idx - i * NCAT1;
  float v;
  if (o < H1) {
    v = root[i * H1 + o];
  } else {
    int r = (o - H1) >> 6;
    int j = (o - H1) & 63;
    float s = 0.f;
#pragma unroll
    for (int b = 0; b < NBASES; ++b)
      s += comp[r * NBASES + b] * basis[((long)b * GDIM + i) * H1 + j];
    v = s;
  }
  W[idx] = v;
}

// Wcat2[64][128]: cols 0..63 = w_rel, cols 64..127 = w_root
__global__ __launch_bounds__(256) void build_wcat2(
    const float* __restrict__ w_rel, const float* __restrict__ w_root,
    float* __restrict__ W) {
  int idx = blockIdx.x * blockDim.x + threadIdx.x;
  if (idx >= H1 * NCAT2) return;
  int i = idx / NCAT2;
  int o = idx - i * NCAT2;
  W[idx] = (o < H2) ? w_rel[i * H2 + o] : w_root[i * H2 + (o - H2)];
}

// ---------------- per-(dst, rel) edge counts ----------------

__global__ __launch_bounds__(256) void zero_i32(int* __restrict__ p, int n) {
  int i = blockIdx.x * blockDim.x + threadIdx.x;
  if (i < n) p[i] = 0;
}

__global__ __launch_bounds__(256) void count_edges(
    const int* __restrict__ ei, const int* __restrict__ etype,
    int* __restrict__ cnt) {
  int e = blockIdx.x * blockDim.x + threadIdx.x;
  if (e >= NE) return;
  int dst = ei[NE + e];
  int r = etype[e];
  atomicAdd(&cnt[dst * NREL + r], 1);
}

// ---------------- f32 WMMA GEMM: C[M,N] = A[M,K] @ B[K,N] ----------------
// Block = 8 waves = 128 rows x 64 cols of C.  The K x 64 B-panel is staged
// once into LDS in a K-paired layout so each B fragment is one ds_load_b64.
// Each wave owns a 16x64 strip: one global A float2 load feeds FOUR
// V_WMMA_F32_16X16X4_F32 issues (4 independent accumulator chains).
// Fragment layouts per CDNA5 ISA (wave32):
//   A (16x4): lane l<16 -> row l, K={k,k+1}; lane 16+l -> row l, K={k+2,k+3}
//   B (4x16): lane l<16 -> col l, K={k,k+1}; lane 16+l -> col l, K={k+2,k+3}
//   C/D:      VGPR v -> rows v / v+8 (lane halves), lane&15 = col
template <int M, int N, int K>
__global__ __launch_bounds__(256) void wmma_gemm_f32_t(
    const float* __restrict__ A, const float* __restrict__ B,
    float* __restrict__ C) {
  __shared__ float Blds[K / 2][64][2];  // [k-pair][col][k&1] -> b64-loadable pairs

  const int tid = threadIdx.x;
  const int lane = tid & 31;
  const int wave = tid >> 5;
  const int ncol0 = blockIdx.y * 64;

  // cooperative stage of B[K][ncol0..ncol0+63]
  for (int idx = tid; idx < K * 64; idx += 256) {
    int kk = idx >> 6;
    int c = idx & 63;
    Blds[kk >> 1][c][kk & 1] = B[kk * N + ncol0 + c];
  }
  __syncthreads();

  const int half = lane >> 4;  // which K-pair this lane carries
  const int l15 = lane & 15;
  const int row0 = blockIdx.x * 128 + wave * 16;

  if (row0 < M) {  // wave-uniform: 16-row tiles divide M exactly
    const float* Arow = A + (long)(row0 + l15) * K;
    v8f acc0 = {}, acc1 = {}, acc2 = {}, acc3 = {};

    for (int k = 0; k < K; k += 4) {
      const int ka = k + (half << 1);
      v2f a = *(const v2f*)(Arow + ka);
      const float* bp = &Blds[ka >> 1][l15][0];
      v2f b0 = *(const v2f*)(bp);
      v2f b1 = *(const v2f*)(bp + 32);
      v2f b2 = *(const v2f*)(bp + 64);
      v2f b3 = *(const v2f*)(bp + 96);
      acc0 = __builtin_amdgcn_wmma_f32_16x16x4_f32(false, a, false, b0,
                                                   (short)0, acc0, false, false);
      acc1 = __builtin_amdgcn_wmma_f32_16x16x4_f32(false, a, false, b1,
                                                   (short)0, acc1, false, false);
      acc2 = __builtin_amdgcn_wmma_f32_16x16x4_f32(false, a, false, b2,
                                                   (short)0, acc2, false, false);
      acc3 = __builtin_amdgcn_wmma_f32_16x16x4_f32(false, a, false, b3,
                                                   (short)0, acc3, false, false);
    }

    float* Cb = C + (long)(row0 + half * 8) * N + ncol0 + l15;
#pragma unroll
    for (int v = 0; v < 8; ++v) {
      Cb[(long)v * N] = acc0[v];
      Cb[(long)v * N + 16] = acc1[v];
      Cb[(long)v * N + 32] = acc2[v];
      Cb[(long)v * N + 48] = acc3[v];
    }
  }
}

// ---------------- layer 1: init + mean-aggregate ----------------

__global__ __launch_bounds__(256) void init_out1(
    const float* __restrict__ Y, const float* __restrict__ bias1,
    float* __restrict__ out1) {
  int idx = blockIdx.x * blockDim.x + threadIdx.x;
  if (idx >= NN * H1) return;
  int n = idx >> 6;
  int j = idx & 63;
  out1[idx] = Y[(long)n * NCAT1 + j] + bias1[j];
}

// out1[dst,j] += Y[src, 64 + 64*r + j] / cnt[dst,r]   (64 lanes per edge)
__global__ __launch_bounds__(256) void agg1(
    const int* __restrict__ ei, const int* __restrict__ etype,
    const int* __restrict__ cnt, const float* __restrict__ Y,
    float* __restrict__ out1) {
  long tid = (long)blockIdx.x * blockDim.x + threadIdx.x;
  long e = tid >> 6;
  if (e >= NE) return;
  int j = (int)(tid & 63);
  int src = ei[e];
  int dst = ei[NE + e];
  int r = etype[e];
  float c = (float)cnt[dst * NREL + r];  // >= 1 since this edge exists
  float v = Y[(long)src * NCAT1 + H1 + (r << 6) + j] / c;
  atomicAdd(&out1[(long)dst * H1 + j], v);
}

// ---------------- layer 2: init + weighted-sum aggregate ----------------

__global__ __launch_bounds__(256) void init_out2(
    const float* __restrict__ Z, const float* __restrict__ b_rel,
    float* __restrict__ out) {
  int idx = blockIdx.x * blockDim.x + threadIdx.x;
  if (idx >= NN * H2) return;
  int n = idx >> 6;
  int j = idx & 63;
  out[idx] = Z[(long)n * NCAT2 + H2 + j] + b_rel[j];  // root part + bias
}

// out[dst,j] += norm[e] * Z[src, j]   (64 lanes per edge; Z cols 0..63 = w_rel part)
__global__ __launch_bounds__(256) void agg2(
    const int* __restrict__ ei, const float* __restrict__ enorm,
    const float* __restrict__ Z, float* __restrict__ out) {
  long tid = (long)blockIdx.x * blockDim.x + threadIdx.x;
  long e = tid >> 6;
  if (e >= NE) return;
  int j = (int)(tid & 63);
  int src = ei[e];
  int dst = ei[NE + e];
  float v = enorm[e] * Z[(long)src * NCAT2 + j];
  atomicAdd(&out[(long)dst * H2 + j], v);
}

// ---------------- launch ----------------

extern "C" void kernel_launch(void* const* d_in, const int* in_sizes, int n_in,
                              void* d_out, int out_size, void* d_ws, size_t ws_size,
                              hipStream_t stream) {
  const float* x      = (const float*)d_in[0];   // [N,128]
  const int*   ei     = (const int*)d_in[1];     // [2,E]
  const float* enorm  = (const float*)d_in[2];   // [E]
  const int*   etype  = (const int*)d_in[3];     // [E]
  const float* basis  = (const float*)d_in[4];   // [30,128,64]
  const float* comp   = (const float*)d_in[5];   // [8,30]
  const float* root   = (const float*)d_in[6];   // [128,64]
  const float* bias1  = (const float*)d_in[7];   // [64]
  const float* w_rel  = (const float*)d_in[8];   // [64,64]
  const float* b_rel  = (const float*)d_in[9];   // [64]
  const float* w_root = (const float*)d_in[10];  // [64,64]
  float* out = (float*)d_out;                    // [N,64]

  // workspace carve-up
  float* Y    = (float*)d_ws;                         // N*576
  float* Z    = Y + (size_t)NN * NCAT1;               // N*128
  float* out1 = Z + (size_t)NN * NCAT2;               // N*64
  int*   cnt  = (int*)(out1 + (size_t)NN * H1);       // N*8
  float* Wc1  = (float*)(cnt + (size_t)NN * NREL);    // 128*576
  float* Wc2  = Wc1 + GDIM * NCAT1;                   // 64*128

  // 1) fold basis decomposition + pack weights
  build_wcat1<<<(GDIM * NCAT1 + 255) / 256, 256, 0, stream>>>(basis, comp, root, Wc1);
  build_wcat2<<<(H1 * NCAT2 + 255) / 256, 256, 0, stream>>>(w_rel, w_root, Wc2);

  // 2) per-(dst,rel) edge counts for the mean
  zero_i32<<<(NN * NREL + 255) / 256, 256, 0, stream>>>(cnt, NN * NREL);
  count_edges<<<(NE + 255) / 256, 256, 0, stream>>>(ei, etype, cnt);

  // 3) GEMM1: Y = X @ [root | W_0 .. W_7]   (100000 x 128 x 576)
  {
    dim3 grid((NN + 127) / 128, NCAT1 / 64);
    wmma_gemm_f32_t<NN, NCAT1, GDIM><<<grid, 256, 0, stream>>>(x, Wc1, Y);
  }

  // 4) layer-1 combine: root part + bias, then per-relation mean aggregation
  init_out1<<<(NN * H1 + 255) / 256, 256, 0, stream>>>(Y, bias1, out1);
  {
    long t = (long)NE * H1;
    agg1<<<(unsigned)((t + 255) / 256), 256, 0, stream>>>(ei, etype, cnt, Y, out1);
  }

  // 5) GEMM2: Z = out1 @ [w_rel | w_root]   (100000 x 64 x 128)
  {
    dim3 grid((NN + 127) / 128, NCAT2 / 64);
    wmma_gemm_f32_t<NN, NCAT2, H1><<<grid, 256, 0, stream>>>(out1, Wc2, Z);
  }

  // 6) layer-2 combine: root part + bias, then edge-weighted sum aggregation
  init_out2<<<(NN * H2 + 255) / 256, 256, 0, stream>>>(Z, b_rel, out);
  {
    long t = (long)NE * H2;
    agg2<<<(unsigned)((t + 255) / 256), 256, 0, stream>>>(ei, enorm, Z, out);
  }
}